// GraphEncoder_10857677324489
// MI455X (gfx1250) — compile-verified
//
#include <hip/hip_runtime.h>

typedef __attribute__((ext_vector_type(2))) float v2f;
typedef __attribute__((ext_vector_type(8))) float v8f;

#define NN 100000
#define NE 1600000

// ---------------- utility kernels ----------------

__global__ void fill_zero_kernel(float* __restrict__ p, int n) {
  int i = blockIdx.x * blockDim.x + threadIdx.x;
  if (i < n) p[i] = 0.0f;
}

__global__ void deg_count_kernel(const long long* __restrict__ dst, float* __restrict__ deg) {
  int e = blockIdx.x * blockDim.x + threadIdx.x;
  if (e < NE) atomicAdd(&deg[(int)dst[e]], 1.0f);
}

__global__ void deg_inv_kernel(float* __restrict__ deg) {
  int i = blockIdx.x * blockDim.x + threadIdx.x;
  if (i < NN) {
    float d = deg[i];
    deg[i] = (d > 0.0f) ? (1.0f / d) : 0.0f;
  }
}

// ---------------- feature-parallel scatter (segment_sum of x[src] into agg[dst]) ----------------
// One thread per (edge, 4-feature group): coalesced float4 read of x[src],
// 4 coalesced global_atomic_add_f32 into agg[dst].
template <int F>
__global__ void scatter_kernel(const float* __restrict__ x,
                               const long long* __restrict__ src,
                               const long long* __restrict__ dst,
                               float* __restrict__ agg) {
  constexpr int G = F / 4;
  long long tid = (long long)blockIdx.x * blockDim.x + threadIdx.x;
  if (tid >= (long long)NE * G) return;
  int e = (int)(tid / G);
  int g = (int)(tid - (long long)e * G);
  int s = (int)src[e];
  int d = (int)dst[e];
  float4 v = ((const float4*)(x + (long long)s * F))[g];
  float* a = agg + (long long)d * F + g * 4;
  atomicAdd(a + 0, v.x);
  atomicAdd(a + 1, v.y);
  atomicAdd(a + 2, v.z);
  atomicAdd(a + 3, v.w);
}

// ---------------- SAGE dense layer via V_WMMA_F32_16X16X4_F32 ----------------
// out[m, n] = relu?( sum_k mean[m,k]*Wl[n,k] + sum_k x[m,k]*Wr[n,k] + b[n] )
// mean[m,k] = agg[m,k] * dinv[m] (folded into the A-fragment load).
// One wave32 block computes a 16-row slab x full OUT width (NT 16-col tiles).
template <int K, int OUT, int NT, bool RELU>
__global__ __launch_bounds__(32) void sage_gemm_kernel(
    const float* __restrict__ agg,   // [NN, K] neighbor sums
    const float* __restrict__ xin,   // [NN, K] self features
    const float* __restrict__ dinv,  // [NN]
    const float* __restrict__ Wl,    // [OUT, K]
    const float* __restrict__ Wr,    // [OUT, K]
    const float* __restrict__ bias,  // [OUT]
    float* __restrict__ out)         // [NN, OUT]
{
  const int lane = threadIdx.x;      // 0..31
  const int half = lane >> 4;        // 0: K=k..k+1, 1: K=k+2..k+3 (A and B layouts)
  const int ml   = lane & 15;        // A row / B column within tile
  const int m0   = blockIdx.x * 16;
  const int row  = m0 + ml;

  const float dv = dinv[row];
  const float2* __restrict__ arow = (const float2*)(agg + (long long)row * K);
  const float2* __restrict__ xrow = (const float2*)(xin + (long long)row * K);

  v8f acc[NT];
#pragma unroll
  for (int t = 0; t < NT; ++t) { v8f z = {}; acc[t] = z; }

  // ---- neighbor-mean path: mean @ Wl.T ----
#pragma unroll
  for (int k = 0; k < K; k += 4) {
    float2 am = arow[(k >> 1) + half];
    v2f a; a.x = am.x * dv; a.y = am.y * dv;
#pragma unroll
    for (int t = 0; t < NT; ++t) {
      int n = t * 16 + ml;
      float2 bw;
      if (OUT >= 16 || n < OUT) bw = ((const float2*)(Wl + (long long)n * K))[(k >> 1) + half];
      else                      bw = make_float2(0.0f, 0.0f);
      v2f b; b.x = bw.x; b.y = bw.y;
      acc[t] = __builtin_amdgcn_wmma_f32_16x16x4_f32(
          /*neg_a=*/false, a, /*neg_b=*/false, b,
          /*c_mod=*/(short)0, acc[t], /*reuse_a=*/false, /*reuse_b=*/false);
    }
  }

  // ---- self path: x @ Wr.T ----
#pragma unroll
  for (int k = 0; k < K; k += 4) {
    float2 xm = xrow[(k >> 1) + half];
    v2f a; a.x = xm.x; a.y = xm.y;
#pragma unroll
    for (int t = 0; t < NT; ++t) {
      int n = t * 16 + ml;
      float2 bw;
      if (OUT >= 16 || n < OUT) bw = ((const float2*)(Wr + (long long)n * K))[(k >> 1) + half];
      else                      bw = make_float2(0.0f, 0.0f);
      v2f b; b.x = bw.x; b.y = bw.y;
      acc[t] = __builtin_amdgcn_wmma_f32_16x16x4_f32(
          false, a, false, b, (short)0, acc[t], false, false);
    }
  }

  // ---- bias + relu + store (D layout: VGPR r -> M = r + 8*half, N = ml) ----
#pragma unroll
  for (int t = 0; t < NT; ++t) {
    int n = t * 16 + ml;
    float bv = (OUT >= 16 || n < OUT) ? bias[n] : 0.0f;
#pragma unroll
    for (int r = 0; r < 8; ++r) {
      float v = acc[t][r] + bv;
      if (RELU) v = fmaxf(v, 0.0f);
      int mr = m0 + r + 8 * half;
      if (OUT >= 16 || n < OUT) out[(long long)mr * OUT + n] = v;
    }
  }
}

// ---------------- launch ----------------

extern "C" void kernel_launch(void* const* d_in, const int* in_sizes, int n_in,
                              void* d_out, int out_size, void* d_ws, size_t ws_size,
                              hipStream_t stream) {
  const float*     x   = (const float*)d_in[0];
  const long long* ei  = (const long long*)d_in[1];
  const float* W1l = (const float*)d_in[2];
  const float* b1  = (const float*)d_in[3];
  const float* W1r = (const float*)d_in[4];
  const float* W2l = (const float*)d_in[5];
  const float* b2  = (const float*)d_in[6];
  const float* W2r = (const float*)d_in[7];
  const float* W4l = (const float*)d_in[8];
  const float* b4  = (const float*)d_in[9];
  const float* W4r = (const float*)d_in[10];
  float* out = (float*)d_out;

  float* ws   = (float*)d_ws;
  float* deg  = ws;                                 // [NN]   degree -> deg_inv (in place)
  float* agg  = deg + NN;                           // [NN*128] neighbor-sum scratch (reused per layer)
  float* h1   = agg + (size_t)NN * 128;             // [NN*64]
  float* h2   = h1  + (size_t)NN * 64;              // [NN*128]

  const long long* src = ei;
  const long long* dst = ei + NE;

  const int TB = 256;
  auto blocks = [](long long n, int tb) { return (int)((n + tb - 1) / tb); };

  // degrees
  fill_zero_kernel<<<blocks(NN, TB), TB, 0, stream>>>(deg, NN);
  deg_count_kernel<<<blocks(NE, TB), TB, 0, stream>>>(dst, deg);
  deg_inv_kernel<<<blocks(NN, TB), TB, 0, stream>>>(deg);

  // ---- layer 1: 40 -> 64, relu ----
  fill_zero_kernel<<<blocks((long long)NN * 40, TB), TB, 0, stream>>>(agg, NN * 40);
  scatter_kernel<40><<<blocks((long long)NE * 10, TB), TB, 0, stream>>>(x, src, dst, agg);
  sage_gemm_kernel<40, 64, 4, true><<<NN / 16, 32, 0, stream>>>(agg, x, deg, W1l, W1r, b1, h1);

  // ---- layer 2: 64 -> 128, relu ----
  fill_zero_kernel<<<blocks((long long)NN * 64, TB), TB, 0, stream>>>(agg, NN * 64);
  scatter_kernel<64><<<blocks((long long)NE * 16, TB), TB, 0, stream>>>(h1, src, dst, agg);
  sage_gemm_kernel<64, 128, 8, true><<<NN / 16, 32, 0, stream>>>(agg, h1, deg, W2l, W2r, b2, h2);

  // ---- layer 3: 128 -> 3, no relu ----
  fill_zero_kernel<<<blocks((long long)NN * 128, TB), TB, 0, stream>>>(agg, NN * 128);
  scatter_kernel<128><<<blocks((long long)NE * 32, TB), TB, 0, stream>>>(h2, src, dst, agg);
  sage_gemm_kernel<128, 3, 1, false><<<NN / 16, 32, 0, stream>>>(agg, h2, deg, W4l, W4r, b4, out);
}